// scattering_59210419142945
// MI455X (gfx1250) — compile-verified
//
#include <hip/hip_runtime.h>
#include <hip/hip_bf16.h>

// ---------------------------------------------------------------------------
// Graph scattering on MI455X (gfx1250).
//
// Math reduction: Lmat in the reference is exactly diagonal
// (diag(deg) * dhalf_i * dhalf_j), so eigh -> permutation eigenvectors and
// g_j = V diag(lam_j) V^T is diagonal with g_j[v,v] = h_j(deg_v/max(1,deg_v)).
// All h_j >= 0, so |.| factors through, and the output is
//   out[0:128]            = (1/V) sum_v f[v,:]
//   out[128+p*128 : +128] = (1/V) sum_v c[p][v] * |f[v,:]|,  p = 0..71
// with c[j][v]=h_j(v), c[8+j1*8+j2][v]=h_{j1}(v)h_{j2}(v).
// => one 73x2048x128 f32 GEMM + a 2048x2048 row-sum.
//
// Roofline: dominant traffic is W (16 MB -> ~0.7 us at 23.3 TB/s). The GEMM
// is only ~38 MFLOP, so it must not become a latency tail: K is split into
// 8 chunks (64 blocks, 384 waves) so each wave's dependent WMMA chain is
// ~32 ops per accumulator; partials are reduced by a tiny final kernel.
// f (1 MB) and c (640 KB) are L2-resident (192 MB L2) -> direct global loads
// into WMMA fragments; no LDS/TDM staging needed.
// ---------------------------------------------------------------------------

#define V_NODES 2048
#define D_F     128
#define P_ROWS  80           // 72 live rows, padded to 5 M-tiles of 16
#define KCHUNKS 8
#define KLEN    (V_NODES / KCHUNKS)   // 256
#define PART_ROWS 81         // rows 0..79 = |f|-GEMM tile rows, row 80 = mean row
#define PART_STRIDE (PART_ROWS * 16)  // floats per (kchunk, ntile) partial
#define OUT_ELEMS 9344       // 128 * (1 + 8 + 64)
#define INV_V   (1.0f / 2048.0f)

typedef __attribute__((ext_vector_type(2))) float v2f;
typedef __attribute__((ext_vector_type(8))) float v8f;

// ---------------------------------------------------------------------------
// Kernel 1: deg[r] = sum_k W[r,k].  One block per row, float4 loads, LDS tree.
// Streams the 16 MB W matrix exactly once (the roofline-critical read).
// ---------------------------------------------------------------------------
__global__ void __launch_bounds__(256) deg_kernel(const float* __restrict__ W,
                                                  float* __restrict__ deg) {
  __shared__ float red[256];
  const float4* row = (const float4*)(W + (size_t)blockIdx.x * V_NODES);
  float s = 0.0f;
  for (int i = threadIdx.x; i < V_NODES / 4; i += 256) {
    float4 v = row[i];
    s += (v.x + v.y) + (v.z + v.w);
  }
  red[threadIdx.x] = s;
  __syncthreads();
  for (int off = 128; off > 0; off >>= 1) {
    if (threadIdx.x < (unsigned)off) red[threadIdx.x] += red[threadIdx.x + off];
    __syncthreads();
  }
  if (threadIdx.x == 0) deg[blockIdx.x] = red[0];
}

// ---------------------------------------------------------------------------
// Kernel 2: per-node filter responses -> padded weight matrix c[80][2048].
//   A = R*ln(2)/(J-R+1) = ln(2)/2;  g_hat(x) = 0.5 - 0.5*cos(2*pi*x/A) on (-A,0]
//   h[0] = sqrt(max(1.125 - sum wav^2, 0)), h[1..7] = wavelets js=2..8.
// ---------------------------------------------------------------------------
__global__ void __launch_bounds__(256) weights_kernel(const float* __restrict__ deg,
                                                      float* __restrict__ c) {
  const int v = blockIdx.x * 256 + threadIdx.x;
  if (v >= V_NODES) return;

  const float kA = 0.34657359027997264f;   // ln(2)/2
  const float kW = 18.129440567308175f;    // 2*pi / kA

  const float d   = deg[v];
  const float lam = d / fmaxf(1.0f, d);    // diagonal of normalized Laplacian
  const float E   = fabsf(lam);
  const float lgE = logf(E);

  float h[8];
  float ssum = 0.0f;
#pragma unroll
  for (int i = 1; i <= 7; ++i) {
    const float x = lgE - kA * (float)i * (1.0f / 3.0f);
    float wv = 0.0f;
    if (x > -kA && x <= 0.0f) wv = 0.5f - 0.5f * cosf(kW * x);
    h[i] = wv;
    ssum += wv * wv;
  }
  h[0] = sqrtf(fmaxf(1.125f - ssum, 0.0f));  // scaling function

  // rows 0..7: layer-1 weights; rows 8..71: layer-2 products; 72..79: zero pad
#pragma unroll
  for (int j = 0; j < 8; ++j) c[(size_t)j * V_NODES + v] = h[j] * INV_V;
#pragma unroll
  for (int j1 = 0; j1 < 8; ++j1)
#pragma unroll
    for (int j2 = 0; j2 < 8; ++j2)
      c[(size_t)(8 + j1 * 8 + j2) * V_NODES + v] = h[j1] * h[j2] * INV_V;
#pragma unroll
  for (int p = 72; p < P_ROWS; ++p) c[(size_t)p * V_NODES + v] = 0.0f;
}

// ---------------------------------------------------------------------------
// Kernel 3: K-split f32 WMMA GEMM -> partial tiles.
//   Grid (8 n-tiles, 8 k-chunks) x 192 threads (6 waves, wave32).
//   waves 0..4: D(16x16) += A(16x4, rows of c) x B(4x16, |f|) over K=256 chunk
//   wave 5:     signed mean row: A row0 = 1/V (constant), B = f.
//   V_WMMA_F32_16X16X4_F32 operand layout (ISA 7.12.2):
//     A: lanes 0-15 hold M=lane, {K=0,K=1}; lanes 16-31 hold M=lane-16, {K=2,K=3}
//     B (mirror): lanes 0-15 hold N=lane, rows {K=0,K=1}; lanes 16-31 rows {K=2,K=3}
//     D: VGPR i -> M = i (lanes 0-15) / i+8 (lanes 16-31), N = lane&15
// ---------------------------------------------------------------------------
__global__ void __launch_bounds__(192) scat_gemm_partial_kernel(
    const float* __restrict__ c, const float* __restrict__ f,
    float* __restrict__ part) {
  const int n0    = blockIdx.x * 16;
  const int kbase = blockIdx.y * KLEN;
  const int wave  = threadIdx.x >> 5;
  const int lane  = threadIdx.x & 31;
  const int l16   = lane & 15;
  const int hi    = lane >> 4;          // 0: K=0,1 half ; 1: K=2,3 half

  float* myPart = part + (size_t)(blockIdx.y * 8 + blockIdx.x) * PART_STRIDE;

  v8f acc0 = {};
  v8f acc1 = {};

  // per-lane B base: column n0+l16, starting row kbase + 2*hi
  const float* fB = f + (size_t)(kbase + 2 * hi) * D_F + n0 + l16;

  if (wave < 5) {
    // ---- |f| GEMM over this K chunk ----
    const int prow = wave * 16 + l16;
    const float* cA = c + (size_t)prow * V_NODES + kbase + 2 * hi;

    for (int k0 = 0; k0 < KLEN; k0 += 8) {
      v2f a0, b0, a1, b1;
      const float2 ca0 = *(const float2*)(cA + k0);
      a0.x = ca0.x; a0.y = ca0.y;
      b0.x = fabsf(fB[(size_t)k0 * D_F]);
      b0.y = fabsf(fB[(size_t)(k0 + 1) * D_F]);
      acc0 = __builtin_amdgcn_wmma_f32_16x16x4_f32(false, a0, false, b0,
                                                   (short)0, acc0, false, false);
      const float2 ca1 = *(const float2*)(cA + k0 + 4);
      a1.x = ca1.x; a1.y = ca1.y;
      b1.x = fabsf(fB[(size_t)(k0 + 4) * D_F]);
      b1.y = fabsf(fB[(size_t)(k0 + 5) * D_F]);
      acc1 = __builtin_amdgcn_wmma_f32_16x16x4_f32(false, a1, false, b1,
                                                   (short)0, acc1, false, false);
    }

#pragma unroll
    for (int i = 0; i < 8; ++i) {
      const int p = wave * 16 + i + hi * 8;     // D-tile row -> global p row
      if (p < 72) myPart[p * 16 + l16] = acc0[i] + acc1[i];
    }
  } else {
    // ---- signed mean row: constant A (M=0 row = 1/V), B = raw f ----
    v2f a;
    a.x = (l16 == 0) ? INV_V : 0.0f;
    a.y = a.x;

    for (int k0 = 0; k0 < KLEN; k0 += 8) {
      v2f b0, b1;
      b0.x = fB[(size_t)k0 * D_F];
      b0.y = fB[(size_t)(k0 + 1) * D_F];
      acc0 = __builtin_amdgcn_wmma_f32_16x16x4_f32(false, a, false, b0,
                                                   (short)0, acc0, false, false);
      b1.x = fB[(size_t)(k0 + 4) * D_F];
      b1.y = fB[(size_t)(k0 + 5) * D_F];
      acc1 = __builtin_amdgcn_wmma_f32_16x16x4_f32(false, a, false, b1,
                                                   (short)0, acc1, false, false);
    }
    if (hi == 0) myPart[80 * 16 + l16] = acc0[0] + acc1[0];   // mean row (M=0)
  }
}

// ---------------------------------------------------------------------------
// Kernel 4: reduce the 8 K-chunk partials into the final 9344-element output.
// Fixed summation order -> deterministic across graph replays.
// ---------------------------------------------------------------------------
__global__ void __launch_bounds__(256) reduce_kernel(const float* __restrict__ part,
                                                     float* __restrict__ out) {
  const int idx = blockIdx.x * 256 + threadIdx.x;
  if (idx >= OUT_ELEMS) return;

  int row, n;
  if (idx < 128) {          // parts[0]: signed mean row
    row = 80;
    n = idx;
  } else {                  // parts[1], parts[2]: |f|-GEMM rows 0..71
    const int t = idx - 128;
    row = t >> 7;           // p
    n = t & 127;            // feature
  }
  const int nt  = n >> 4;
  const int col = n & 15;

  float s = 0.0f;
#pragma unroll
  for (int kc = 0; kc < KCHUNKS; ++kc)
    s += part[(size_t)(kc * 8 + nt) * PART_STRIDE + row * 16 + col];
  out[idx] = s;
}

// ---------------------------------------------------------------------------
// Host launcher.
// Workspace: deg[2048] + c[80*2048] + part[64*81*16] floats  (~973 KB).
// ---------------------------------------------------------------------------
extern "C" void kernel_launch(void* const* d_in, const int* in_sizes, int n_in,
                              void* d_out, int out_size, void* d_ws, size_t ws_size,
                              hipStream_t stream) {
  (void)in_sizes; (void)n_in; (void)out_size; (void)ws_size;

  const float* W = (const float*)d_in[0];   // [2048, 2048] f32
  const float* f = (const float*)d_in[1];   // [2048, 128]  f32
  float* out = (float*)d_out;               // [9344] f32

  float* deg  = (float*)d_ws;               // [2048]
  float* c    = deg + V_NODES;              // [80 * 2048]
  float* part = c + (size_t)P_ROWS * V_NODES;  // [64 * 81 * 16]

  deg_kernel<<<V_NODES, 256, 0, stream>>>(W, deg);
  weights_kernel<<<V_NODES / 256, 256, 0, stream>>>(deg, c);
  dim3 ggrid(D_F / 16, KCHUNKS);
  scat_gemm_partial_kernel<<<ggrid, 192, 0, stream>>>(c, f, part);
  reduce_kernel<<<(OUT_ELEMS + 255) / 256, 256, 0, stream>>>(part, out);
}